// MultiLayerSuperGATConvNet_36515811950910
// MI455X (gfx1250) — compile-verified
//
#include <hip/hip_runtime.h>
#include <hip/hip_bf16.h>

typedef __attribute__((ext_vector_type(2))) float v2f;
typedef __attribute__((ext_vector_type(8))) float v8f;

#define NEG_SLOPE 0.2f
#define EPS_SM 1e-16f

__device__ __forceinline__ float wave_sum32(float v) {
#pragma unroll
  for (int off = 16; off > 0; off >>= 1) v += __shfl_xor(v, off, 32);
  return v;
}

// Float atomic max via signed/unsigned int ordering trick (works for all signs,
// init value -inf). Lowers to GLOBAL_ATOMIC_MAX_I32 / GLOBAL_ATOMIC_MIN_U32.
__device__ __forceinline__ void atomicMaxF(float* addr, float val) {
  if (val >= 0.0f) atomicMax((int*)addr, __float_as_int(val));
  else             atomicMin((unsigned int*)addr, __float_as_uint(val));
}

// ---------------------------------------------------------------------------
// C[M,128] = A[M,K] @ B[K,128], fp32 via V_WMMA_F32_16X16X4_F32.
// Block = 16 rows x 128 cols: 8 waves, each owns one 16x16 output tile.
// A strip (16 x K) staged in LDS with +4 float row padding (ldl % 64 == 4
// => the 16 same-column reads by lanes 0..15 land in distinct banks).
// A frag (16x4 f32): lanes 0-15 row M=lane hold K {k,k+1}; lanes 16-31 {k+2,k+3}.
// B frag (4x16 f32): lane column N=lane&15; v0/v1 hold K {k,k+1} (lo lanes)
// or {k+2,k+3} (hi lanes). C/D: VGPR i -> row i + 8*(lane>=16), col lane&15.
// ---------------------------------------------------------------------------
__global__ __launch_bounds__(256) void gemm_wmma_f32(
    const float* __restrict__ A, const float* __restrict__ B,
    float* __restrict__ C, int M, int K) {
  __shared__ float lds[16 * 260];  // supports K <= 256
  const int ldl  = K + 4;
  const int row0 = blockIdx.x * 16;
  const int tid  = threadIdx.x;
  const int kc4  = K >> 2;  // float4 per row
  for (int i = tid; i < (kc4 << 4); i += 256) {
    int r = i / kc4, c = i - r * kc4;
    float4 v = *(const float4*)(A + (size_t)(row0 + r) * K + (c << 2));
    float* p = &lds[r * ldl + (c << 2)];
    p[0] = v.x; p[1] = v.y; p[2] = v.z; p[3] = v.w;
  }
  __syncthreads();

  const int lane = tid & 31;
  const int n0   = (tid >> 5) << 4;   // wave's 16-column tile
  const int mr   = lane & 15;
  const int kh   = (lane >> 4) << 1;  // 0 (lanes 0-15) or 2 (lanes 16-31)
  v8f acc = {};
  for (int k = 0; k < K; k += 4) {
    v2f a, b;
    a.x = lds[mr * ldl + k + kh];
    a.y = lds[mr * ldl + k + kh + 1];
    b.x = B[(size_t)(k + kh)     * 128 + n0 + mr];
    b.y = B[(size_t)(k + kh + 1) * 128 + n0 + mr];
    acc = __builtin_amdgcn_wmma_f32_16x16x4_f32(
        false, a, false, b, (short)0, acc, false, false);
  }
  const int rbase = row0 + ((lane >> 4) << 3);
#pragma unroll
  for (int i = 0; i < 8; ++i) {
    int r = rbase + i;
    if (r < M) C[(size_t)r * 128 + n0 + mr] = acc[i];
  }
}

// sl[n] = h[n,:]·att_l ; sr[n] = h[n,:]·att_r   (F = 128, one wave per node)
__global__ __launch_bounds__(256) void node_scores(
    const float* __restrict__ h, const float* __restrict__ al,
    const float* __restrict__ ar, float* __restrict__ sl,
    float* __restrict__ sr, int N) {
  int wid  = (int)(((long long)blockIdx.x * 256 + threadIdx.x) >> 5);
  int lane = threadIdx.x & 31;
  if (wid >= N) return;
  float4 hv = *(const float4*)(h + (size_t)wid * 128 + lane * 4);
  float4 av = *(const float4*)(al + lane * 4);
  float4 rv = *(const float4*)(ar + lane * 4);
  float pl = hv.x*av.x + hv.y*av.y + hv.z*av.z + hv.w*av.w;
  float pr = hv.x*rv.x + hv.y*rv.y + hv.z*rv.z + hv.w*rv.w;
  pl = wave_sum32(pl); pr = wave_sum32(pr);
  if (lane == 0) { sl[wid] = pl; sr[wid] = pr; }
}

__global__ __launch_bounds__(256) void fill_f(float* __restrict__ p,
                                              long long n, float v) {
  for (long long i = (long long)blockIdx.x * 256 + threadIdx.x; i < n;
       i += (long long)gridDim.x * 256)
    p[i] = v;
}

// One wave per edge: MX logit = h[s]·h[d] (wave reduce); alpha = (sl+sr)*sigmoid;
// leaky_relu; record + running segment max.
__global__ __launch_bounds__(256) void edge_alpha128(
    const float* __restrict__ h, const float* __restrict__ sl,
    const float* __restrict__ sr, const int* __restrict__ srcp,
    const int* __restrict__ dstp, long long E, long long E2,
    float* __restrict__ alE, float* __restrict__ amax) {
  long long wid = ((long long)blockIdx.x * 256 + threadIdx.x) >> 5;
  int lane = threadIdx.x & 31;
  if (wid >= E2) return;
  int s, d;
  if (wid < E) { s = srcp[wid]; d = dstp[wid]; } else { s = d = (int)(wid - E); }
  float4 a = *(const float4*)(h + (size_t)s * 128 + lane * 4);
  float4 b = *(const float4*)(h + (size_t)d * 128 + lane * 4);
  float p = a.x*b.x + a.y*b.y + a.z*b.z + a.w*b.w;
  p = wave_sum32(p);
  if (lane == 0) {
    float sig   = 1.0f / (1.0f + expf(-p));
    float alpha = (sl[s] + sr[d]) * sig;
    alpha = alpha > 0.0f ? alpha : NEG_SLOPE * alpha;
    alE[wid] = alpha;
    atomicMaxF(&amax[d], alpha);
  }
}

// Thread per edge: e = exp(alpha - amax[dst]); denom[dst] += e (in-place alE).
__global__ __launch_bounds__(256) void edge_expsum(
    float* __restrict__ alE, const float* __restrict__ amax,
    float* __restrict__ dnm, const int* __restrict__ dstp,
    long long E, long long E2) {
  long long e = (long long)blockIdx.x * 256 + threadIdx.x;
  if (e >= E2) return;
  int d = (e < E) ? dstp[e] : (int)(e - E);
  float v = expf(alE[e] - amax[d]);
  alE[e] = v;
  atomicAdd(&dnm[d], v);
}

// One wave per edge: out[dst,:] += h[src,:] * (e / (denom + eps)).
__global__ __launch_bounds__(256) void edge_scatter128(
    const float* __restrict__ h, const float* __restrict__ alE,
    const float* __restrict__ dnm, const int* __restrict__ srcp,
    const int* __restrict__ dstp, long long E, long long E2,
    float* __restrict__ out) {
  long long wid = ((long long)blockIdx.x * 256 + threadIdx.x) >> 5;
  int lane = threadIdx.x & 31;
  if (wid >= E2) return;
  int s, d;
  if (wid < E) { s = srcp[wid]; d = dstp[wid]; } else { s = d = (int)(wid - E); }
  float coef = alE[wid] / (dnm[d] + EPS_SM);
  float4 a = *(const float4*)(h + (size_t)s * 128 + lane * 4);
  float* o = out + (size_t)d * 128 + lane * 4;
  atomicAdd(o + 0, a.x * coef);
  atomicAdd(o + 1, a.y * coef);
  atomicAdd(o + 2, a.z * coef);
  atomicAdd(o + 3, a.w * coef);
}

__global__ __launch_bounds__(256) void bias_act(
    float* __restrict__ p, const float* __restrict__ b,
    long long total, int F, int relu) {
  for (long long i = (long long)blockIdx.x * 256 + threadIdx.x; i < total;
       i += (long long)gridDim.x * 256) {
    float v = p[i] + b[i % F];
    p[i] = relu ? fmaxf(v, 0.0f) : v;
  }
}

// h3[n] = h[n,:128] · w[:,0]   (one wave per node)
__global__ __launch_bounds__(256) void gemv128(
    const float* __restrict__ h, const float* __restrict__ w,
    float* __restrict__ out, int N) {
  int wid  = (int)(((long long)blockIdx.x * 256 + threadIdx.x) >> 5);
  int lane = threadIdx.x & 31;
  if (wid >= N) return;
  float4 a = *(const float4*)(h + (size_t)wid * 128 + lane * 4);
  float4 b = *(const float4*)(w + lane * 4);
  float p = a.x*b.x + a.y*b.y + a.z*b.z + a.w*b.w;
  p = wave_sum32(p);
  if (lane == 0) out[wid] = p;
}

// Scalar-feature (F=1) variants for layer 3.
__global__ __launch_bounds__(256) void edge_alpha_s(
    const float* __restrict__ h3, const float* __restrict__ al,
    const float* __restrict__ ar, const int* __restrict__ srcp,
    const int* __restrict__ dstp, long long E, long long E2,
    float* __restrict__ alE, float* __restrict__ amax) {
  long long e = (long long)blockIdx.x * 256 + threadIdx.x;
  if (e >= E2) return;
  int s, d;
  if (e < E) { s = srcp[e]; d = dstp[e]; } else { s = d = (int)(e - E); }
  float hs = h3[s], hd = h3[d];
  float sig   = 1.0f / (1.0f + expf(-hs * hd));
  float alpha = (hs * al[0] + hd * ar[0]) * sig;
  alpha = alpha > 0.0f ? alpha : NEG_SLOPE * alpha;
  alE[e] = alpha;
  atomicMaxF(&amax[d], alpha);
}

__global__ __launch_bounds__(256) void edge_scatter_s(
    const float* __restrict__ h3, const float* __restrict__ alE,
    const float* __restrict__ dnm, const int* __restrict__ srcp,
    const int* __restrict__ dstp, long long E, long long E2,
    float* __restrict__ out) {
  long long e = (long long)blockIdx.x * 256 + threadIdx.x;
  if (e >= E2) return;
  int s, d;
  if (e < E) { s = srcp[e]; d = dstp[e]; } else { s = d = (int)(e - E); }
  float coef = alE[e] / (dnm[d] + EPS_SM);
  atomicAdd(&out[d], h3[s] * coef);
}

extern "C" void kernel_launch(void* const* d_in, const int* in_sizes, int n_in,
                              void* d_out, int out_size, void* d_ws, size_t ws_size,
                              hipStream_t stream) {
  (void)n_in; (void)out_size; (void)ws_size;
  const float* x   = (const float*)d_in[0];
  const int*   ei  = (const int*)d_in[1];   // [2, E]
  const float* W1  = (const float*)d_in[2];
  const float* al1 = (const float*)d_in[3];
  const float* ar1 = (const float*)d_in[4];
  const float* b1  = (const float*)d_in[5];
  const float* W2  = (const float*)d_in[6];
  const float* al2 = (const float*)d_in[7];
  const float* ar2 = (const float*)d_in[8];
  const float* b2  = (const float*)d_in[9];
  const float* W3  = (const float*)d_in[10];
  const float* al3 = (const float*)d_in[11];
  const float* ar3 = (const float*)d_in[12];
  const float* b3  = (const float*)d_in[13];

  const int       N  = in_sizes[0] / 256;
  const long long E  = in_sizes[1] / 2;
  const long long E2 = E + N;
  const int* srcp = ei;
  const int* dstp = ei + E;

  // Workspace carve-out (floats). ~56 MB total.
  float* ws   = (float*)d_ws;
  float* ht   = ws;                       // [N,128] transformed features
  float* agg  = ht  + (size_t)N * 128;    // [N,128] aggregation / next input
  float* sl   = agg + (size_t)N * 128;    // [N]
  float* sr   = sl  + N;                  // [N]
  float* amax = sr  + N;                  // [N]
  float* dnm  = amax + N;                 // [N]
  float* alE  = dnm + N;                  // [E2]
  float* h3   = alE + E2;                 // [N]

  const dim3 blk(256);
  const unsigned gGemm  = (unsigned)((N + 15) / 16);
  const unsigned gNodeW = (unsigned)((N + 7) / 8);          // wave per node
  const unsigned gEdgeW = (unsigned)((E2 + 7) / 8);         // wave per edge
  const unsigned gEdgeT = (unsigned)((E2 + 255) / 256);     // thread per edge
  const unsigned gFeat  = (unsigned)(((long long)N * 128 + 255) / 256);
  const unsigned gNodeT = (unsigned)((N + 255) / 256);
  const float NEG_INF = -__builtin_inff();

  // One 128-feature SuperGAT layer. `in` is only read by the GEMM, so `agg`
  // may alias `in` (stream ordering: GEMM issues before agg is zeroed).
  auto layer128 = [&](const float* in, int K, const float* W, const float* al,
                      const float* ar, const float* b, int relu) {
    gemm_wmma_f32<<<gGemm, blk, 0, stream>>>(in, W, ht, N, K);
    node_scores  <<<gNodeW, blk, 0, stream>>>(ht, al, ar, sl, sr, N);
    fill_f<<<gFeat,  blk, 0, stream>>>(agg, (long long)N * 128, 0.0f);
    fill_f<<<gNodeT, blk, 0, stream>>>(amax, N, NEG_INF);
    fill_f<<<gNodeT, blk, 0, stream>>>(dnm, N, 0.0f);
    edge_alpha128 <<<gEdgeW, blk, 0, stream>>>(ht, sl, sr, srcp, dstp, E, E2, alE, amax);
    edge_expsum   <<<gEdgeT, blk, 0, stream>>>(alE, amax, dnm, dstp, E, E2);
    edge_scatter128<<<gEdgeW, blk, 0, stream>>>(ht, alE, dnm, srcp, dstp, E, E2, agg);
    bias_act<<<gFeat, blk, 0, stream>>>(agg, b, (long long)N * 128, 128, relu);
  };

  layer128(x,   256, W1, al1, ar1, b1, 1);   // agg = relu(layer1)
  layer128(agg, 128, W2, al2, ar2, b2, 1);   // agg = relu(layer2)

  // Layer 3: F_out = 1 (scalar per node), output into d_out.
  float* outp = (float*)d_out;
  gemv128<<<gNodeW, blk, 0, stream>>>(agg, W3, h3, N);
  fill_f<<<gNodeT, blk, 0, stream>>>(outp, N, 0.0f);
  fill_f<<<gNodeT, blk, 0, stream>>>(amax, N, NEG_INF);
  fill_f<<<gNodeT, blk, 0, stream>>>(dnm, N, 0.0f);
  edge_alpha_s  <<<gEdgeT, blk, 0, stream>>>(h3, al3, ar3, srcp, dstp, E, E2, alE, amax);
  edge_expsum   <<<gEdgeT, blk, 0, stream>>>(alE, amax, dnm, dstp, E, E2);
  edge_scatter_s<<<gEdgeT, blk, 0, stream>>>(h3, alE, dnm, srcp, dstp, E, E2, outp);
  bias_act<<<gNodeT, blk, 0, stream>>>(outp, b3, N, 1, 0);
}